// My_conv2d_609885356902
// MI455X (gfx1250) — compile-verified
//
#include <hip/hip_runtime.h>

typedef __attribute__((ext_vector_type(2))) float v2f;
typedef __attribute__((ext_vector_type(4))) float v4f;
typedef __attribute__((ext_vector_type(8))) float v8f;

#define UNITS   64
#define DEPTH   128
#define CHANNEL 33
#define TRAIN_W (DEPTH + 1)   // 129 floats per train row
#define NEIGH_W 96            // 1 id + 63 pad + 32 slots
#define SLOT_OFF UNITS        // slots start at col 64
#define OUT_W   (UNITS + 1)   // 65 floats per output row

// Pre-swizzled B: [4 n-tiles][32 k-steps][32 lanes][2 floats] = 8192 floats = 32 KB
#define BPACK_ELEMS (4 * 32 * 32 * 2)

// B lane layout for V_WMMA_F32_16X16X4_F32 (4x16 K-major, mirrors 32-bit A):
// lanes 0-15 hold (K=4k+0, 4k+1) for N=n0+lane; lanes 16-31 hold (K=4k+2, 4k+3).
__global__ void build_bpack(const float* __restrict__ kern, float* __restrict__ bp) {
  int f = blockIdx.x * blockDim.x + threadIdx.x;
  if (f >= BPACK_ELEMS) return;
  int j    = f & 1;          // which of the 2 floats in the lane's b64
  int lane = (f >> 1) & 31;
  int k    = (f >> 6) & 31;  // k-step (4 K values each)
  int tile = f >> 11;        // n-tile
  int lo = lane & 15, hi = lane >> 4;
  int u = tile * 16 + lo;            // output unit (column of out)
  int d = 4 * k + 2 * hi + j;        // depth index
  // w[u][d] = kernel[u][d][CHANNEL-1]
  bp[f] = kern[(u * DEPTH + d) * CHANNEL + (CHANNEL - 1)];
}

__device__ __forceinline__ int lower_bound_strided(const float* __restrict__ base,
                                                   int stride, int n, float key) {
  int lo = 0, hi = n;
  while (lo < hi) {
    int mid = (lo + hi) >> 1;
    if (base[(long)mid * stride] < key) lo = mid + 1; else hi = mid;
  }
  return lo;
}

__global__ void __launch_bounds__(128)
fused_searchsorted_gemm(const float* __restrict__ train,
                        const float* __restrict__ neigh,
                        const float* __restrict__ bpack,
                        float* __restrict__ out, int L) {
  __shared__ float yL[16][132];        // fused A operand, padded stride (bank-conflict free)
  __shared__ float Bp[BPACK_ELEMS];    // 32 KB swizzled B
  __shared__ float sL[16];
  __shared__ int   ppL[16];
  __shared__ int   npL[16];

  const int tid = threadIdx.x;
  const int rowBase = blockIdx.x * 16;
  if (rowBase >= L) return;

  // Stage swizzled B into LDS (32 KB, served from L2 after first touch).
  {
    const v4f* __restrict__ src = (const v4f*)bpack;
    v4f* dst = (v4f*)Bp;
    #pragma unroll 4
    for (int i = tid; i < BPACK_ELEMS / 4; i += 128) dst[i] = src[i];
  }

  // Binary searches (searchsorted, side='left') for this block's 16 rows.
  if (tid < 16) {
    sL[tid] = 0.0f;
    const int gr = rowBase + tid;
    const float key = train[(long)gr * TRAIN_W];            // train id
    ppL[tid] = lower_bound_strided(neigh, NEIGH_W, L, key); // into neigh ids
    npL[tid] = lower_bound_strided(train, TRAIN_W, L, key); // into train ids
  }
  __syncthreads();

  // s[i] = sum_j (neigh[pp[i], 64+j] == -1 ? j : 0); 8 threads/row, 4 slots each.
  {
    const int r = tid >> 3;
    const int part = tid & 7;
    const long base = (long)ppL[r] * NEIGH_W + SLOT_OFF;
    float partial = 0.0f;
    #pragma unroll
    for (int j = 0; j < 4; ++j) {
      const int sj = part * 4 + j;
      const float v = neigh[base + sj];
      if (v == -1.0f) partial += (float)sj;
    }
    atomicAdd(&sL[r], partial);   // ds_add_f32
  }
  __syncthreads();

  // Fused A operand: y[r][d] = train[gr][1+d] + s_r * train[np_r][d], d = 0..127.
  // Thread t handles column d = t for all 16 rows -> fully coalesced row loads.
  for (int r = 0; r < 16; ++r) {
    const long gr = rowBase + r;
    const float sr = sL[r];
    const long np = npL[r];
    yL[r][tid] = train[gr * TRAIN_W + 1 + tid] + sr * train[np * TRAIN_W + tid];
  }
  __syncthreads();

  // Index column.
  if (tid < 16) {
    out[(long)(rowBase + tid) * OUT_W] = (float)(rowBase + tid);
  }

  // Each wave owns one 16x16 output tile (16 rows x 16 units), K=128 in 32 steps.
  const int wave = tid >> 5;
  const int lane = tid & 31;
  const int lo = lane & 15;
  const int hi = lane >> 4;

  v8f acc = {0.f, 0.f, 0.f, 0.f, 0.f, 0.f, 0.f, 0.f};
  const float* __restrict__ brow = &Bp[wave * 2048 + lane * 2];
  #pragma unroll 8
  for (int k = 0; k < 32; ++k) {
    // A: 16x4 f32 tile; lane lo reads y[lo][4k + 2*hi .. +1] as one b64.
    v2f a = *(const v2f*)&yL[lo][4 * k + 2 * hi];
    // B: contiguous per-lane b64 from the swizzled store.
    v2f b = *(const v2f*)&brow[k * 64];
    acc = __builtin_amdgcn_wmma_f32_16x16x4_f32(
        /*neg_a=*/false, a, /*neg_b=*/false, b,
        /*c_mod=*/(short)0, acc, /*reuse_a=*/false, /*reuse_b=*/false);
  }

  // C/D layout: VGPR r -> row r (lanes 0-15) / row 8+r (lanes 16-31), col = lo.
  const int n = wave * 16 + lo;
  #pragma unroll
  for (int r8 = 0; r8 < 8; ++r8) {
    const int row = r8 + 8 * hi;
    out[(long)(rowBase + row) * OUT_W + 1 + n] = acc[r8];
  }
}

extern "C" void kernel_launch(void* const* d_in, const int* in_sizes, int n_in,
                              void* d_out, int out_size, void* d_ws, size_t ws_size,
                              hipStream_t stream) {
  const float* train = (const float*)d_in[0];   // (1, L, 129) f32
  const float* neigh = (const float*)d_in[1];   // (1, L, 96)  f32
  const float* kern  = (const float*)d_in[2];   // (64, 128, 33) f32
  float* out   = (float*)d_out;                 // (L, 65) f32
  float* bpack = (float*)d_ws;                  // 32 KB scratch

  const int L = in_sizes[0] / TRAIN_W;

  build_bpack<<<(BPACK_ELEMS + 255) / 256, 256, 0, stream>>>(kern, bpack);
  fused_searchsorted_gemm<<<(L + 15) / 16, 128, 0, stream>>>(train, neigh, bpack, out, L);
}